// Yolov4Loss_23252952941259
// MI455X (gfx1250) — compile-verified
//
#include <hip/hip_runtime.h>
#include <math.h>

#define B_   16
#define L_   60
#define NC_  80
#define NA_  3
#define NH_  76
#define NW_  76
#define NCH_ 85
#define CH_        (NH_*NW_)        /* 5776  */
#define CELLS_PB_  (NA_*CH_)        /* 17328 */
#define TOTAL_CELLS_ (B_*CELLS_PB_) /* 277248 */
#define EPS_ 1e-16f
#define IGNORE_ 0.5f

typedef float v2f __attribute__((ext_vector_type(2)));
typedef float v8f __attribute__((ext_vector_type(8)));

__device__ __forceinline__ float clip_logf(float x) { return fmaxf(logf(x), -100.0f); }
__device__ __forceinline__ float sigm(float z)      { return 1.0f / (1.0f + expf(-z)); }
__device__ __forceinline__ float bcef(float p, float t) {
    return -(t * clip_logf(p) + (1.0f - t) * clip_logf(1.0f - p));
}

// ---------------- K0: zero workspace ----------------
__global__ void k_zero(int* cellLab, double* acc) {
    int idx = blockIdx.x * blockDim.x + threadIdx.x;
    if (idx < TOTAL_CELLS_) cellLab[idx] = 0;
    if (idx == 0) *acc = 0.0;
}

// ---------------- K1: per-label prep ----------------
__global__ void k_prep(const float* __restrict__ tgt, float* __restrict__ recF,
                       int* __restrict__ recI, int* __restrict__ nvalid) {
    int b = blockIdx.x;
    int l = threadIdx.x;
    __shared__ int flags[64];
    __shared__ int nv;
    float cf = 0.f, x = 0.f, y = 0.f, w = 0.f, h = 0.f;
    if (l < L_) {
        const float* p = tgt + (size_t)(b * L_ + l) * 5;
        cf = p[0]; x = p[1]; y = p[2]; w = p[3]; h = p[4];
        flags[l] = ((cf + x + y + w + h) > 0.0f) ? 1 : 0;
    } else if (l < 64) {
        flags[l] = 0;
    }
    __syncthreads();
    if (l == 0) {
        int s = 0;
        for (int k = 0; k < L_; ++k) s += flags[k];
        nv = s;
        nvalid[b] = s;
    }
    __syncthreads();
    if (l < L_) {
        float tx = x * NW_, ty = y * NH_, tw = w * NW_, th = h * NH_;
        const float AW[3] = {1.25f, 2.0f, 4.125f};
        const float AH[3] = {1.625f, 3.75f, 2.875f};
        int bn = 0; float bestc = -1e30f;
        for (int a = 0; a < 3; ++a) {
            float awk = AW[a], ahk = AH[a];
            float miw = fminf(tw, awk), mih = fminf(th, ahk);
            float en  = (miw > 0.0f && mih > 0.0f) ? 1.0f : 0.0f;
            float ai  = miw * mih * en;
            float au  = tw * th + awk * ahk - ai;
            float iou = ai / fmaxf(au, EPS_);
            float dw = tw - awk, dh = th - ahk;
            float rho2 = 0.25f * (dw * dw + dh * dh);
            float cw = fmaxf(tw, awk), chh = fmaxf(th, ahk);
            float c2 = cw * cw + chh * chh + EPS_;
            float dv = atanf(tw / fmaxf(th, EPS_)) - atanf(awk / fmaxf(ahk, EPS_));
            float v  = 0.40528473456935109f * dv * dv;   // 4/pi^2
            float alpha = v / fmaxf(1.0f - iou + v, EPS_);
            float ciou  = iou - (rho2 / c2 + v * alpha);
            if (ciou > bestc) { bestc = ciou; bn = a; }   // first-tie argmax
        }
        int ti = (int)tx, tj = (int)ty;
        float dx = tx - floorf(tx), dy = ty - floorf(ty);
        float lw = logf(tw / AW[bn] + EPS_), lh = logf(th / AH[bn] + EPS_);
        float scale = sqrtf(2.0f - tw * th * (1.0f / (float)(NW_ * NH_)));
        int sel = (l < nv) ? 1 : 0;   // valid & best_mask (best_mask always true, NA==3)
        float* rf = recF + (size_t)(b * L_ + l) * 12;
        rf[0] = tx; rf[1] = ty; rf[2] = tw; rf[3] = th;
        rf[4] = dx; rf[5] = dy; rf[6] = lw; rf[7] = lh; rf[8] = scale;
        int* ri = recI + (size_t)(b * L_ + l) * 8;
        ri[0] = ti; ri[1] = tj; ri[2] = bn; ri[3] = sel; ri[4] = (int)cf;
    }
}

// ---------------- K2: deterministic scatter (last writer wins) ----------------
__global__ void k_scatter(const int* __restrict__ recI, int* __restrict__ cellLab) {
    int b = threadIdx.x;
    if (b >= B_) return;
    for (int l = 0; l < L_; ++l) {
        const int* ri = recI + (size_t)(b * L_ + l) * 8;
        if (ri[3]) {
            int ti = ri[0], tj = ri[1], bn = ri[2];
            if (ti >= 0 && ti < NW_ && tj >= 0 && tj < NH_ && bn >= 0 && bn < NA_)
                cellLab[((b * NA_ + bn) * NH_ + tj) * NW_ + ti] = l + 1;
        }
    }
}

// ---------------- K3: main loss ----------------
__global__ void k_loss(const float* __restrict__ out, const float* __restrict__ recF,
                       const int* __restrict__ recI, const int* __restrict__ nvalid,
                       const int* __restrict__ cellLab, double* __restrict__ acc) {
    __shared__ float s_tb[L_ * 4];
    __shared__ int   s_nv;
    int b   = blockIdx.y;
    int tid = threadIdx.x;
    if (tid == 0) {
        __builtin_prefetch(recF + (size_t)b * L_ * 12, 0, 1);  // global_prefetch_b8
        s_nv = nvalid[b];
    }
    if (tid < L_ * 4) s_tb[tid] = recF[(size_t)(b * L_ + (tid >> 2)) * 12 + (tid & 3)];
    __syncthreads();

    int n = blockIdx.x * blockDim.x + tid;
    float loss = 0.0f;
    if (n < CELLS_PB_) {
        int a = n / CH_;
        int rem = n - a * CH_;
        int j = rem / NW_;
        int i = rem - j * NW_;
        const float* op = out + ((size_t)(b * NA_ + a) * NCH_) * CH_ + (size_t)j * NW_ + i;
        float z0 = op[0];
        float z1 = op[(size_t)CH_];
        float z2 = op[2 * (size_t)CH_];
        float z3 = op[3 * (size_t)CH_];
        float z4 = op[4 * (size_t)CH_];
        float p0 = sigm(z0), p1 = sigm(z1), p4 = sigm(z4);
        const float AW[3] = {1.25f, 2.0f, 4.125f};
        const float AH[3] = {1.625f, 3.75f, 2.875f};
        float px = p0 + (float)i, py = p1 + (float)j;
        float pw = expf(z2) * AW[a], ph = expf(z3) * AH[a];
        float hpw = 0.5f * pw, hph = 0.5f * ph;
        float paw = pw * ph;
        int nv = s_nv;
        float best = -INFINITY;
        for (int l = 0; l < nv; ++l) {
            float tx = s_tb[l * 4 + 0], ty = s_tb[l * 4 + 1];
            float tw = s_tb[l * 4 + 2], th = s_tb[l * 4 + 3];
            float tlx = fmaxf(px - hpw, tx - 0.5f * tw);
            float brx = fminf(px + hpw, tx + 0.5f * tw);
            float tly = fmaxf(py - hph, ty - 0.5f * th);
            float bry = fminf(py + hph, ty + 0.5f * th);
            float en  = (tlx < brx && tly < bry) ? 1.0f : 0.0f;
            float ai  = (brx - tlx) * (bry - tly) * en;
            float au  = paw + tw * th - ai;
            float iou = ai / fmaxf(au, EPS_);
            best = fmaxf(best, iou);
        }
        int cl = cellLab[b * CELLS_PB_ + n];
        float obj = (cl > 0 || nv == 0) ? 1.0f : ((best <= IGNORE_) ? 1.0f : 0.0f);
        if (cl > 0) {
            int l = cl - 1;
            const float* rf = recF + (size_t)(b * L_ + l) * 12;
            const int*   ri = recI + (size_t)(b * L_ + l) * 8;
            float dx = rf[4], dy = rf[5], lw = rf[6], lh = rf[7], sc = rf[8];
            int cls = ri[4];
            float w2 = sc * sc;
            loss += w2 * (bcef(p0, dx) + bcef(p1, dy));            // loss_xy
            float d2 = (z2 - lw) * sc, d3 = (z3 - lh) * sc;
            loss += 0.5f * (d2 * d2 + d3 * d3);                    // loss_wh
            loss += -clip_logf(p4);                                // loss_obj, t=1
            for (int c = 0; c < NC_; ++c) {                        // loss_cls
                float p = sigm(op[(size_t)(5 + c) * CH_]);
                loss += bcef(p, (c == cls) ? 1.0f : 0.0f);
            }
        } else {
            // only obj channel contributes (all masks/scales are zero here)
            if (obj > 0.5f) loss += -clip_logf(1.0f - p4);
        }
    }

    // Wave-level sum via WMMA: D = A * ones(4x16); column of D (lanes 0 & 16)
    // holds the 16 row-sums -> lane0's 8 + lane16's 8 D-regs sum to the wave total.
    v2f av; av[0] = loss; av[1] = 0.0f;
    v2f bv; bv[0] = 1.0f; bv[1] = 1.0f;
    v8f cv = {};
    v8f dv = __builtin_amdgcn_wmma_f32_16x16x4_f32(false, av, false, bv,
                                                   (short)0, cv, false, false);
    float s = dv[0] + dv[1] + dv[2] + dv[3] + dv[4] + dv[5] + dv[6] + dv[7];
    float tot = __shfl(s, 0, 32) + __shfl(s, 16, 32);
    if ((tid & 31) == 0) atomicAdd(acc, (double)tot);
}

// ---------------- K4: finalize ----------------
__global__ void k_final(const double* __restrict__ acc, float* __restrict__ outp) {
    if (blockIdx.x == 0 && threadIdx.x == 0) outp[0] = (float)(*acc / (double)B_);
}

extern "C" void kernel_launch(void* const* d_in, const int* in_sizes, int n_in,
                              void* d_out, int out_size, void* d_ws, size_t ws_size,
                              hipStream_t stream) {
    (void)in_sizes; (void)n_in; (void)out_size; (void)ws_size;
    const float* output = (const float*)d_in[0];
    const float* target = (const float*)d_in[1];
    float* outp = (float*)d_out;

    char* ws = (char*)d_ws;
    float*  recF    = (float*)(ws + 0);                  // 16*60*12*4 = 46080 B
    int*    recI    = (int*)  (ws + 46080);              // 16*60*8*4  = 30720 B
    int*    nvalid  = (int*)  (ws + 76800);              // 256 B slot
    int*    cellLab = (int*)  (ws + 77056);              // 277248*4 = 1108992 B
    double* acc     = (double*)(ws + 1186048);           // 8 B

    k_zero<<<(TOTAL_CELLS_ + 255) / 256, 256, 0, stream>>>(cellLab, acc);
    k_prep<<<B_, 64, 0, stream>>>(target, recF, recI, nvalid);
    k_scatter<<<1, 32, 0, stream>>>(recI, cellLab);
    dim3 g3((CELLS_PB_ + 255) / 256, B_);
    k_loss<<<g3, 256, 0, stream>>>(output, recF, recI, nvalid, cellLab, acc);
    k_final<<<1, 32, 0, stream>>>(acc, outp);
}